// RegularSimilar_47227460387300
// MI455X (gfx1250) — compile-verified
//
#include <hip/hip_runtime.h>
#include <math.h>

typedef float v2f __attribute__((ext_vector_type(2)));
typedef float v8f __attribute__((ext_vector_type(8)));

#define DIMS 64
#define SNUM 500

// ---------------------------------------------------------------------------
// Kernel 1: uif = union_feature (N x 128) @ W^T (128 x 64) + b via f32 WMMA.
// One wave (32 lanes) computes one 16x16 output tile; K=128 in steps of 4
// using V_WMMA_F32_16X16X4_F32.
//   A layout (16x4): lanes 0-15 -> M=lane, K={k0,k0+1}; lanes 16-31 -> K={k0+2,k0+3}
//   B layout (4x16): mirrored K split across lane halves, N = lane&15
//   D layout (16x16): vgpr r -> M = r + 8*(lane>=16), N = lane&15
// ---------------------------------------------------------------------------
__global__ __launch_bounds__(32) void uif_gemm_wmma(
    const float* __restrict__ UF,   // (N, 128)
    const float* __restrict__ W,    // (64, 128), we need W^T
    const float* __restrict__ bias, // (64)
    float* __restrict__ OUT)        // (N, 64)
{
    const int lane = threadIdx.x & 31;
    const int half = lane >> 4;     // 0: low lanes, 1: high lanes
    const int l    = lane & 15;
    const int ntn  = DIMS / 16;     // 4 tiles along N
    const int m0   = (blockIdx.x / ntn) * 16;
    const int n0   = (blockIdx.x % ntn) * 16;

    v8f acc = {};
    for (int k0 = 0; k0 < 2 * DIMS; k0 += 4) {
        const int kb = k0 + 2 * half;
        v2f a, b;
        a.x = UF[(size_t)(m0 + l) * (2 * DIMS) + kb + 0];
        a.y = UF[(size_t)(m0 + l) * (2 * DIMS) + kb + 1];
        // B[k][n] = W^T[k][n] = W[n][k]
        b.x = W[(size_t)(n0 + l) * (2 * DIMS) + kb + 0];
        b.y = W[(size_t)(n0 + l) * (2 * DIMS) + kb + 1];
        acc = __builtin_amdgcn_wmma_f32_16x16x4_f32(
            /*neg_a=*/false, a, /*neg_b=*/false, b,
            /*c_mod=*/(short)0, acc, /*reuse_a=*/false, /*reuse_b=*/false);
    }

    const int col = n0 + l;
    const float bb = bias[col];
#pragma unroll
    for (int r = 0; r < 8; ++r) {
        const int row = m0 + r + half * 8;
        OUT[(size_t)row * DIMS + col] = acc[r] + bb;
    }
}

// ---------------------------------------------------------------------------
// Kernel 2: zero the two scalar accumulators (deterministic per launch).
// ---------------------------------------------------------------------------
__global__ void init_scalars(float* scalars) {
    if (threadIdx.x < 2) scalars[threadIdx.x] = 0.0f;
}

// ---------------------------------------------------------------------------
// Kernel 3: per-row fused pipeline. One 256-thread block (8 waves) per row.
// ---------------------------------------------------------------------------
__global__ __launch_bounds__(256) void regsim_main(
    const int*   __restrict__ need_replace, // (N,2)
    const float* __restrict__ all_items,    // (n_items, 64)
    const int*   __restrict__ sample_items, // (n_users, S)
    const float* __restrict__ gumbel_u,     // (N, S)
    const float* __restrict__ uif,          // (N, 64) from workspace
    int*         __restrict__ out_items,    // (N) int32 bit patterns
    float*       __restrict__ out_feat,     // (N, 64)
    float*       __restrict__ out_scalars,  // [loss, mean_sim]
    int N)
{
    __shared__ float emb[DIMS];
    __shared__ float uvec[DIMS];
    __shared__ float rs[SNUM];
    __shared__ float key[SNUM];
    __shared__ int   sid[SNUM];
    __shared__ float bv[256];
    __shared__ int   bi[256];
    __shared__ int   s_kstar;
    __shared__ int   s_jstar;
    __shared__ float s_sim;

    const int i = blockIdx.x;
    const int t = threadIdx.x;
    const int uid = need_replace[2 * i + 0];
    const int iid = need_replace[2 * i + 1];

    if (t < DIMS) {
        emb[t]  = all_items[(size_t)iid * DIMS + t];
        uvec[t] = uif[(size_t)i * DIMS + t];
    }
    __syncthreads();

    // Fused dual dot products + gumbel key over the gathered sample features.
    for (int j = t; j < SNUM; j += 256) {
        const int s = sample_items[(size_t)uid * SNUM + j];
        sid[j] = s;
        const float4* fp = (const float4*)(all_items + (size_t)s * DIMS);
        float r = 0.0f, p = 0.0f;
#pragma unroll
        for (int q = 0; q < DIMS / 4; ++q) {
            const float4 f = fp[q];
            r = fmaf(emb[4*q+0], f.x, r);  r = fmaf(emb[4*q+1], f.y, r);
            r = fmaf(emb[4*q+2], f.z, r);  r = fmaf(emb[4*q+3], f.w, r);
            p = fmaf(uvec[4*q+0], f.x, p); p = fmaf(uvec[4*q+1], f.y, p);
            p = fmaf(uvec[4*q+2], f.z, p); p = fmaf(uvec[4*q+3], f.w, p);
        }
        rs[j] = r;
        const float eps = 1e-7f;
        const float u = gumbel_u[(size_t)i * SNUM + j];
        const float g = -logf(-logf(u * (1.0f - 2.0f * eps) + eps));
        key[j] = p + g;  // argmax of softmax((p+g)/tau) == argmax of (p+g)
    }
    __syncthreads();

    // Argmax over key with first-index tie break (matches jnp.argmax).
    float bvv = -INFINITY;
    int   bii = SNUM;
    for (int j = t; j < SNUM; j += 256) {
        const float v = key[j];
        if (v > bvv || (v == bvv && j < bii)) { bvv = v; bii = j; }
    }
    bv[t] = bvv; bi[t] = bii;
    __syncthreads();
    if (t == 0) {
        float bb = bv[0]; int ib = bi[0];
        for (int q = 1; q < 256; ++q) {
            if (bv[q] > bb || (bv[q] == bb && bi[q] < ib)) { bb = bv[q]; ib = bi[q]; }
        }
        s_kstar = ib;
    }
    __syncthreads();
    const int kstar = s_kstar;

    // Stable ascending rank of each j; find rank(kstar) and the inverse j*.
    for (int j = t; j < SNUM; j += 256) {
        const float v = rs[j];
        int rank = 0;
        for (int l = 0; l < SNUM; ++l) {
            const float w = rs[l];
            rank += (int)((w < v) || (w == v && l < j));
        }
        if (rank == kstar) s_jstar = j;           // ranks form a permutation
        if (j == kstar)    s_sim  = (float)(rank + 1) / (float)SNUM;
    }
    __syncthreads();

    // Outputs.
    const int kid = sid[kstar];
    if (t < DIMS) out_feat[(size_t)i * DIMS + t] = all_items[(size_t)kid * DIMS + t];
    if (t == 0) {
        out_items[i] = sid[s_jstar];
        const float sim = s_sim;
        const float invN = 1.0f / (float)N;
        atomicAdd(&out_scalars[0], fabsf(sim - 0.5f) * invN);
        atomicAdd(&out_scalars[1], sim * invN);
    }
}

// ---------------------------------------------------------------------------
extern "C" void kernel_launch(void* const* d_in, const int* in_sizes, int n_in,
                              void* d_out, int out_size, void* d_ws, size_t ws_size,
                              hipStream_t stream) {
    const int*   need_replace  = (const int*)d_in[0];   // (N,2)
    const float* union_feature = (const float*)d_in[1]; // (N,128)
    const float* all_items     = (const float*)d_in[2]; // (n_items,64)
    const int*   sample_items  = (const int*)d_in[3];   // (n_users,S)
    const float* W             = (const float*)d_in[4]; // (64,128)
    const float* bias          = (const float*)d_in[5]; // (64)
    const float* gumbel_u      = (const float*)d_in[6]; // (N,S)

    const int N = in_sizes[0] / 2;

    float* uif = (float*)d_ws;                          // N*64 floats scratch

    float* out_f      = (float*)d_out;
    int*   out_items  = (int*)d_out;                    // first N slots: int32 bits
    float* out_feat   = out_f + N;                      // N*64 floats
    float* out_scal   = out_f + N + (size_t)N * DIMS;   // [loss, mean_sim]

    // 1) uif GEMM via f32 WMMA: (N/16)*(64/16) waves, 1 wave per block.
    uif_gemm_wmma<<<(N / 16) * (DIMS / 16), 32, 0, stream>>>(
        union_feature, W, bias, uif);

    // 2) zero scalar accumulators.
    init_scalars<<<1, 32, 0, stream>>>(out_scal);

    // 3) fused per-row pipeline: one 8-wave workgroup per row.
    regsim_main<<<N, 256, 0, stream>>>(
        need_replace, all_items, sample_items, gumbel_u, uif,
        out_items, out_feat, out_scal, N);
}